// Attention_20933670601301
// MI455X (gfx1250) — compile-verified
//
#include <hip/hip_runtime.h>
#include <hip/hip_bf16.h>
#include <math.h>

// Problem dims (fixed by the reference).
#define BB 256   // batch
#define NN 512   // rows per batch
#define CC 256   // channels

typedef float v2f __attribute__((ext_vector_type(2)));
typedef float v8f __attribute__((ext_vector_type(8)));

__device__ __forceinline__ v8f wmma_f32_k4(v2f a, v2f b, v8f c) {
  // D = A(16x4) * B(4x16) + C, full fp32 via V_WMMA_F32_16X16X4_F32
  return __builtin_amdgcn_wmma_f32_16x16x4_f32(false, a, false, b, (short)0, c,
                                               false, false);
}

// ---------------------------------------------------------------------------
// Kernel 1: Q = gather(P, id) @ Wq^T + bq       [256 x 256]
// One wave per 16x16 output tile; K loop steps 4 (f32 WMMA).
// A[m][k] = P[m, id[m], k]   B[k][n] = Wq[n][k]
// ---------------------------------------------------------------------------
__global__ __launch_bounds__(256) void proj_q_kernel(
    const float* __restrict__ P, const int* __restrict__ idx,
    const float* __restrict__ Wq, const float* __restrict__ bq,
    float* __restrict__ Q) {
  const int lane = threadIdx.x & 31;
  const int wave = threadIdx.x >> 5;
  const int tile = blockIdx.x * 8 + wave;       // 256 tiles total
  const int tm = (tile >> 4) << 4;
  const int tn = (tile & 15) << 4;
  const int half = lane >> 4;                   // 0: K+0/1, 1: K+2/3
  const int lm = lane & 15;
  const int mg = tm + lm;                       // A row handled by this lane
  const int ng = tn + lm;                       // B column handled by this lane

  const float* arow = P + ((size_t)mg * NN + (size_t)idx[mg]) * CC;
  const float* brow = Wq + (size_t)ng * CC;     // row n of Wq == column n of B

  v8f acc = {0.f, 0.f, 0.f, 0.f, 0.f, 0.f, 0.f, 0.f};
  for (int kb = 0; kb < CC; kb += 4) {
    const int ks = kb + half * 2;
    v2f a; a.x = arow[ks]; a.y = arow[ks + 1];
    v2f b; b.x = brow[ks]; b.y = brow[ks + 1];
    acc = wmma_f32_k4(a, b, acc);
  }
  const float bias = bq[ng];
  const int mbase = tm + half * 8;              // D: VGPR r -> M = r + 8*half
  #pragma unroll
  for (int r = 0; r < 8; ++r)
    Q[(size_t)(mbase + r) * CC + ng] = acc[r] + bias;
}

// ---------------------------------------------------------------------------
// Kernel 2: W2 = Q @ Wk                          [256 x 256]
// B[k][n] = Wk[k][n] (no transpose).
// ---------------------------------------------------------------------------
__global__ __launch_bounds__(256) void proj_w_kernel(
    const float* __restrict__ Q, const float* __restrict__ Wk,
    float* __restrict__ W2) {
  const int lane = threadIdx.x & 31;
  const int wave = threadIdx.x >> 5;
  const int tile = blockIdx.x * 8 + wave;
  const int tm = (tile >> 4) << 4;
  const int tn = (tile & 15) << 4;
  const int half = lane >> 4;
  const int lm = lane & 15;
  const int mg = tm + lm;
  const int ng = tn + lm;

  const float* arow = Q + (size_t)mg * CC;

  v8f acc = {0.f, 0.f, 0.f, 0.f, 0.f, 0.f, 0.f, 0.f};
  for (int kb = 0; kb < CC; kb += 4) {
    const int ks = kb + half * 2;
    v2f a; a.x = arow[ks]; a.y = arow[ks + 1];
    v2f b;
    b.x = Wk[(size_t)ks * CC + ng];
    b.y = Wk[(size_t)(ks + 1) * CC + ng];
    acc = wmma_f32_k4(a, b, acc);
  }
  const int mbase = tm + half * 8;
  #pragma unroll
  for (int r = 0; r < 8; ++r)
    W2[(size_t)(mbase + r) * CC + ng] = acc[r];
}

// ---------------------------------------------------------------------------
// Kernel 3: the bandwidth kernel. One workgroup per batch. Streams P[b]
// exactly once: scores[n] = P[b,n,:].w_b + c_b, online softmax, and
// T[b,:] = softmax(scores) @ P[b]  (normalized), all fused.
// ---------------------------------------------------------------------------
__global__ __launch_bounds__(256) void attn_stream_kernel(
    const float* __restrict__ P, const float* __restrict__ Q,
    const float* __restrict__ W2, const float* __restrict__ bk,
    float* __restrict__ T) {
  const int b = blockIdx.x;
  const int tid = threadIdx.x;
  const int lane = tid & 31;
  const int wave = tid >> 5;

  __shared__ float s_red[256];
  __shared__ float s_m[8];
  __shared__ float s_l[8];
  __shared__ float s_t[8 * CC];

  // c_b = dot(Q[b,:], bk)  -- scalar offset for all scores of this batch.
  s_red[tid] = Q[(size_t)b * CC + tid] * bk[tid];
  __syncthreads();
  for (int s = 128; s > 0; s >>= 1) {
    if (tid < s) s_red[tid] += s_red[tid + s];
    __syncthreads();
  }
  const float cscal = s_red[0];

  // Per-lane slice of w_b: channels [8*lane, 8*lane+8).
  float w[8];
  {
    const float* wrow = W2 + (size_t)b * CC + lane * 8;
    #pragma unroll
    for (int j = 0; j < 8; ++j) w[j] = wrow[j];
  }

  float m = -INFINITY;
  float l = 0.f;
  float t[8];
  #pragma unroll
  for (int j = 0; j < 8; ++j) t[j] = 0.f;

  const float* Pb = P + (size_t)b * NN * CC;
  for (int i = 0; i < NN / 8; ++i) {
    const int n = wave + 8 * i;                     // this wave's row
    const float* prow = Pb + (size_t)n * CC + lane * 8;
    if (i + 1 < NN / 8) __builtin_prefetch(prow + 8 * CC, 0, 0);

    const float4 pa = *(const float4*)(prow);       // global_load_b128
    const float4 pc = *(const float4*)(prow + 4);   // global_load_b128
    float p[8] = {pa.x, pa.y, pa.z, pa.w, pc.x, pc.y, pc.z, pc.w};

    float part = 0.f;
    #pragma unroll
    for (int j = 0; j < 8; ++j) part = fmaf(p[j], w[j], part);
    #pragma unroll
    for (int off = 16; off >= 1; off >>= 1)
      part += __shfl_xor(part, off, 32);            // wave32 reduce
    const float score = part + cscal;

    // Online softmax update.
    const float mnew = fmaxf(m, score);
    const float scale = __expf(m - mnew);           // 0 on first iteration
    const float e = __expf(score - mnew);
    l = l * scale + e;
    #pragma unroll
    for (int j = 0; j < 8; ++j) t[j] = fmaf(t[j], scale, e * p[j]);
    m = mnew;
  }

  // Merge the 8 per-wave partial softmax states.
  if (lane == 0) { s_m[wave] = m; s_l[wave] = l; }
  #pragma unroll
  for (int j = 0; j < 8; ++j) s_t[wave * CC + lane * 8 + j] = t[j];
  __syncthreads();

  float M = s_m[0];
  #pragma unroll
  for (int wv = 1; wv < 8; ++wv) M = fmaxf(M, s_m[wv]);
  float L = 0.f, tv = 0.f;
  #pragma unroll
  for (int wv = 0; wv < 8; ++wv) {
    const float f = __expf(s_m[wv] - M);
    L += s_l[wv] * f;
    tv = fmaf(s_t[wv * CC + tid], f, tv);
  }
  T[(size_t)b * CC + tid] = tv / L;
}

// ---------------------------------------------------------------------------
// Kernel 4: Out = T @ Wv^T + bv                  [256 x 256]
// B[k][n] = Wv[n][k].
// ---------------------------------------------------------------------------
__global__ __launch_bounds__(256) void proj_out_kernel(
    const float* __restrict__ T, const float* __restrict__ Wv,
    const float* __restrict__ bv, float* __restrict__ Out) {
  const int lane = threadIdx.x & 31;
  const int wave = threadIdx.x >> 5;
  const int tile = blockIdx.x * 8 + wave;
  const int tm = (tile >> 4) << 4;
  const int tn = (tile & 15) << 4;
  const int half = lane >> 4;
  const int lm = lane & 15;
  const int mg = tm + lm;
  const int ng = tn + lm;

  const float* arow = T + (size_t)mg * CC;
  const float* brow = Wv + (size_t)ng * CC;

  v8f acc = {0.f, 0.f, 0.f, 0.f, 0.f, 0.f, 0.f, 0.f};
  for (int kb = 0; kb < CC; kb += 4) {
    const int ks = kb + half * 2;
    v2f a; a.x = arow[ks]; a.y = arow[ks + 1];
    v2f b; b.x = brow[ks]; b.y = brow[ks + 1];
    acc = wmma_f32_k4(a, b, acc);
  }
  const float bias = bv[ng];
  const int mbase = tm + half * 8;
  #pragma unroll
  for (int r = 0; r < 8; ++r)
    Out[(size_t)(mbase + r) * CC + ng] = acc[r] + bias;
}

// ---------------------------------------------------------------------------
extern "C" void kernel_launch(void* const* d_in, const int* in_sizes, int n_in,
                              void* d_out, int out_size, void* d_ws, size_t ws_size,
                              hipStream_t stream) {
  const float* P  = (const float*)d_in[0];
  const int*   id = (const int*)d_in[1];
  const float* Wq = (const float*)d_in[2];
  const float* bq = (const float*)d_in[3];
  const float* Wk = (const float*)d_in[4];
  const float* bk = (const float*)d_in[5];
  const float* Wv = (const float*)d_in[6];
  const float* bv = (const float*)d_in[7];
  float* out = (float*)d_out;

  float* wsQ  = (float*)d_ws;          // [BB, CC]
  float* wsW2 = wsQ  + BB * CC;        // [BB, CC]
  float* wsT  = wsW2 + BB * CC;        // [BB, CC]

  // Q = gather(P, id) @ Wq^T + bq
  proj_q_kernel<<<32, 256, 0, stream>>>(P, id, Wq, bq, wsQ);
  // W2 = Q @ Wk  (so scores = P @ W2_b + Q_b.bk)
  proj_w_kernel<<<32, 256, 0, stream>>>(wsQ, Wk, wsW2);
  // Single streaming pass over P: fused scores + online softmax + weighted sum
  attn_stream_kernel<<<BB, 256, 0, stream>>>(P, wsQ, wsW2, bk, wsT);
  // Out = T @ Wv^T + bv
  proj_out_kernel<<<32, 256, 0, stream>>>(wsT, Wv, bv, out);
}